// SGConvolution_27496380629008
// MI455X (gfx1250) — compile-verified
//
#include <hip/hip_runtime.h>
#include <stdint.h>

#define NW   8     // waves per block in SpMM kernel
#define WAVE 32    // gfx1250 is wave32
#define DF   128   // feature width (reference D_FEAT)

// ---------------------------------------------------------------------------
// 1) degree histogram
// ---------------------------------------------------------------------------
__global__ void k_hist(const int* __restrict__ row, int E, int* __restrict__ deg) {
    int i = blockIdx.x * blockDim.x + threadIdx.x;
    int stride = gridDim.x * blockDim.x;
    for (; i < E; i += stride) atomicAdd(&deg[row[i]], 1);
}

// ---------------------------------------------------------------------------
// 2) exclusive scan of degrees -> row_ptr[N+1]; also seeds cursor[] with the
//    same exclusive prefix (deg and cursor may alias: each element is read
//    into a register by its owning thread before being overwritten).
// ---------------------------------------------------------------------------
__global__ void k_scan(const int* deg, int n, int* row_ptr, int* cursor) {
    __shared__ int s[1024];
    const int tid = threadIdx.x;
    int carry = 0;
    for (int base = 0; base < n; base += 1024) {
        const int i = base + tid;
        const int v = (i < n) ? deg[i] : 0;
        s[tid] = v;
        __syncthreads();
        for (int off = 1; off < 1024; off <<= 1) {
            int y = (tid >= off) ? s[tid - off] : 0;
            __syncthreads();
            s[tid] += y;
            __syncthreads();
        }
        const int incl = s[tid];
        if (i < n) {
            row_ptr[i] = carry + incl - v;
            cursor[i]  = carry + incl - v;
        }
        const int total = s[1023];
        __syncthreads();           // protect s[] before next iteration's write
        carry += total;
    }
    if (tid == 0) row_ptr[n] = carry;
}

// ---------------------------------------------------------------------------
// 3) scatter COO -> CSR using per-row cursors
// ---------------------------------------------------------------------------
__global__ void k_scatter(const int* __restrict__ row, const int* __restrict__ col,
                          const float* __restrict__ val, int E,
                          int* __restrict__ cursor,
                          int* __restrict__ ccol, float* __restrict__ cval) {
    int i = blockIdx.x * blockDim.x + threadIdx.x;
    int stride = gridDim.x * blockDim.x;
    for (; i < E; i += stride) {
        const int r = row[i];
        const int p = atomicAdd(&cursor[r], 1);
        ccol[p] = col[i];
        cval[p] = val[i];
    }
}

// ---------------------------------------------------------------------------
// 4) pull-mode SpMM: one wave per row, lane l owns features [4l, 4l+4).
//    (col,val) chunks of 32 edges are double-buffered into per-wave LDS
//    slices via gfx1250 async-to-LDS loads (ASYNCcnt), overlapping the edge
//    stream with the f32 gather+FMA work. No float atomics anywhere.
// ---------------------------------------------------------------------------
__global__ __launch_bounds__(NW * WAVE) void k_spmm(
        const int* __restrict__ row_ptr,
        const int* __restrict__ ccol, const float* __restrict__ cval,
        const float* __restrict__ xin, float* __restrict__ xout, int n) {
    __shared__ int   sCol[NW][2][WAVE];
    __shared__ float sVal[NW][2][WAVE];

    const int lane = threadIdx.x & (WAVE - 1);
    const int w    = threadIdx.x / WAVE;

    const uint64_t colBase = (uint64_t)(uintptr_t)ccol;
    const uint64_t valBase = (uint64_t)(uintptr_t)cval;
    // LDS byte offsets for this lane's slot in each buffer (addrspace(3)
    // offset == low 32 bits of the generic pointer on amdgcn).
    const uint32_t ldsCol0 = (uint32_t)(uintptr_t)&sCol[w][0][lane];
    const uint32_t ldsCol1 = (uint32_t)(uintptr_t)&sCol[w][1][lane];
    const uint32_t ldsVal0 = (uint32_t)(uintptr_t)&sVal[w][0][lane];
    const uint32_t ldsVal1 = (uint32_t)(uintptr_t)&sVal[w][1][lane];

    const int nWaves = gridDim.x * NW;
    for (int r = blockIdx.x * NW + w; r < n; r += nWaves) {
        const int start = row_ptr[r];
        const int end   = row_ptr[r + 1];
        float4 acc = make_float4(0.f, 0.f, 0.f, 0.f);

        if (end > start) {
            const int nch = (end - start + WAVE - 1) / WAVE;
            // prologue: stage chunk 0 into buffer 0
            {
                int idx = start + lane; if (idx >= end) idx = end - 1;
                const uint32_t off = (uint32_t)idx << 2;
                asm volatile("global_load_async_to_lds_b32 %0, %1, %2"
                             :: "v"(ldsCol0), "v"(off), "s"(colBase) : "memory");
                asm volatile("global_load_async_to_lds_b32 %0, %1, %2"
                             :: "v"(ldsVal0), "v"(off), "s"(valBase) : "memory");
            }
            int buf = 0;
            for (int k = 0; k < nch; ++k) {
                const bool more = (k + 1) < nch;
                if (more) {
                    // stage chunk k+1 into the other buffer, then wait until
                    // only those 2 newest async ops remain outstanding
                    // (async loads complete in order -> chunk k is ready).
                    int idx = start + (k + 1) * WAVE + lane; if (idx >= end) idx = end - 1;
                    const uint32_t off = (uint32_t)idx << 2;
                    const uint32_t dc = buf ? ldsCol0 : ldsCol1;
                    const uint32_t dv = buf ? ldsVal0 : ldsVal1;
                    asm volatile("global_load_async_to_lds_b32 %0, %1, %2"
                                 :: "v"(dc), "v"(off), "s"(colBase) : "memory");
                    asm volatile("global_load_async_to_lds_b32 %0, %1, %2"
                                 :: "v"(dv), "v"(off), "s"(valBase) : "memory");
                    asm volatile("s_wait_asynccnt 0x2" ::: "memory");
                } else {
                    asm volatile("s_wait_asynccnt 0x0" ::: "memory");
                }

                const int cnt = min(WAVE, end - (start + k * WAVE));
                const int*   cbp = &sCol[w][buf][0];
                const float* vbp = &sVal[w][buf][0];
                for (int j = 0; j < cnt; ++j) {
                    const int   c = cbp[j];   // same-address LDS broadcast
                    const float v = vbp[j];
                    const float4 xv =
                        *((const float4*)(xin + ((size_t)c << 7)) + lane);
                    acc.x = fmaf(v, xv.x, acc.x);
                    acc.y = fmaf(v, xv.y, acc.y);
                    acc.z = fmaf(v, xv.z, acc.z);
                    acc.w = fmaf(v, xv.w, acc.w);
                }
                buf ^= 1;
            }
        }
        *((float4*)(xout + ((size_t)r << 7)) + lane) = acc;
    }
}

// ---------------------------------------------------------------------------
// launch: CSR build once, then two pull-SpMM hops (x -> xtmp -> d_out)
// ---------------------------------------------------------------------------
extern "C" void kernel_launch(void* const* d_in, const int* in_sizes, int n_in,
                              void* d_out, int out_size, void* d_ws, size_t ws_size,
                              hipStream_t stream) {
    const float* x     = (const float*)d_in[0];
    const int*   arow  = (const int*)d_in[1];
    const int*   acol  = (const int*)d_in[2];
    const float* avals = (const float*)d_in[3];
    const int E = in_sizes[1];
    const int N = in_sizes[0] / DF;

    // carve scratch out of d_ws (256B-aligned regions), ~39 MB total
    char* ws = (char*)d_ws;
    size_t off = 0;
    auto carve = [&](size_t bytes) -> void* {
        void* p = ws + off;
        off = (off + bytes + 255) & ~(size_t)255;
        return p;
    };
    float* xtmp    = (float*)carve((size_t)N * DF * sizeof(float));
    int*   row_ptr = (int*)  carve((size_t)(N + 1) * sizeof(int));
    int*   cursor  = (int*)  carve((size_t)N * sizeof(int));
    int*   ccol    = (int*)  carve((size_t)E * sizeof(int));
    float* cval    = (float*)carve((size_t)E * sizeof(float));

    const int tpb = 256;
    const int eblocks = (E + tpb - 1) / tpb;

    // CSR build (cursor doubles as degree buffer; memset is capture-safe)
    hipMemsetAsync(cursor, 0, (size_t)N * sizeof(int), stream);
    k_hist<<<eblocks, tpb, 0, stream>>>(arow, E, cursor);
    k_scan<<<1, 1024, 0, stream>>>(cursor, N, row_ptr, cursor);
    k_scatter<<<eblocks, tpb, 0, stream>>>(arow, acol, avals, E, cursor, ccol, cval);

    // two propagation hops
    const int sblocks = (N + NW - 1) / NW;
    k_spmm<<<sblocks, NW * WAVE, 0, stream>>>(row_ptr, ccol, cval, x, xtmp, N);
    k_spmm<<<sblocks, NW * WAVE, 0, stream>>>(row_ptr, ccol, cval, xtmp, (float*)d_out, N);
}